// HeteAttenLayer_11553462026816
// MI455X (gfx1250) — compile-verified
//
#include <hip/hip_runtime.h>
#include <hip/hip_bf16.h>

// ---------------------------------------------------------------------------
// HeteAttenLayer on MI455X (gfx1250, wave32, WMMA).
// All GEMM-shaped work mapped to V_WMMA_F32_16X16X4_F32 tiles (fp32 matches
// the fp32 reference). Intermediates live in d_ws (L2-resident, ~7 MB).
// LDS staging uses the CDNA5 async global->LDS pipe (ASYNCcnt).
// ---------------------------------------------------------------------------

typedef float v2f __attribute__((ext_vector_type(2)));
typedef float v8f __attribute__((ext_vector_type(8)));

#define Bb 16
#define Nn 128
#define Dd 128
#define NEGV (-9.0e15f)
#define LRS 0.2f

__device__ __forceinline__ v8f wmma4(v2f a, v2f b, v8f c) {
    // D = A(16x4 f32) * B(4x16 f32) + C(16x16 f32)
    return __builtin_amdgcn_wmma_f32_16x16x4_f32(false, a, false, b, (short)0, c,
                                                 false, false);
}

__device__ __forceinline__ float lrelu(float x) { return x > 0.f ? x : LRS * x; }

// CDNA5 async copy: 16B global -> LDS, tracked by ASYNCcnt (no VGPR data).
// LDS flat addresses truncate to the LDS byte offset (ISA 10.2), so the low
// 32 bits of the generic shared pointer are a valid LDS destination.
__device__ __forceinline__ void asyncCopy16(void* ldsDst, const void* gsrc) {
    unsigned dst = (unsigned)(uintptr_t)ldsDst;
    unsigned long long src = (unsigned long long)(uintptr_t)gsrc;
    asm volatile("global_load_async_to_lds_b128 %0, %1, off"
                 :
                 : "v"(dst), "v"(src)
                 : "memory");
}
__device__ __forceinline__ void asyncWait0() {
    asm volatile("s_wait_asynccnt 0x0" ::: "memory");
}

// ---------------------------------------------------------------------------
// Kernel 1: typed Q projection.  q[row] = sum_t [mask==t] * (x[row] @ W_Q[t])
// Grid: 128 blocks (one 16-row tile of the 2048 rows), 256 threads (8 waves),
// wave w owns column tile c0 = 16*w. 16x128 A-panel staged in LDS (async).
// ---------------------------------------------------------------------------
__global__ __launch_bounds__(256) void k_qproj(const float* __restrict__ hidden,
                                               const int* __restrict__ ntm,
                                               const float* __restrict__ WQ,
                                               float* __restrict__ q) {
    __shared__ float As[16 * Dd];
    const int row0 = blockIdx.x * 16;

    {
        const char* src = (const char*)(hidden + (size_t)row0 * Dd);
        for (int i = threadIdx.x; i < 16 * Dd * 4 / 16; i += 256)
            asyncCopy16((char*)As + i * 16, src + (size_t)i * 16);
        // prefetch the streamed weight panels into GL2
        __builtin_prefetch(WQ + (size_t)1 * Dd * Dd, 0, 3);
        __builtin_prefetch(WQ + (size_t)2 * Dd * Dd, 0, 3);
        __builtin_prefetch(WQ + (size_t)3 * Dd * Dd, 0, 3);
        asyncWait0();
    }
    __syncthreads();

    const int wave = threadIdx.x >> 5;
    const int lane = threadIdx.x & 31;
    const int nsel = lane & 15;  // A-row within tile / B-col within tile
    const int hi   = lane >> 4;  // upper half-wave -> K offset +2 / M offset +8
    const int c0   = wave * 16;
    const int myType = ntm[row0 + nsel];

    v8f acc = {};
    for (int t = 1; t <= 3; ++t) {
        const float scale = (myType == t) ? 1.f : 0.f;
        const float* W = WQ + (size_t)t * Dd * Dd;
#pragma unroll 8
        for (int kk = 0; kk < 32; ++kk) {
            const int kb = kk * 4 + hi * 2;
            v2f af; af.x = As[nsel * Dd + kb] * scale;
                    af.y = As[nsel * Dd + kb + 1] * scale;
            v2f bf; bf.x = W[(size_t)kb * Dd + c0 + nsel];
                    bf.y = W[(size_t)(kb + 1) * Dd + c0 + nsel];
            acc = wmma4(af, bf, acc);
        }
    }
#pragma unroll
    for (int v = 0; v < 8; ++v) {
        const int r = row0 + v + hi * 8;
        q[(size_t)r * Dd + c0 + nsel] = acc[v];
    }
}

// ---------------------------------------------------------------------------
// Kernel 2: per-batch score matrices.  One block per batch, q[b] in LDS.
// For each 16x16 tile: scross = q@q^T plus 4 homo scores (q .* a_r) @ q^T
// fused in one K-loop (5 accumulators).  Epilogue: tanh(scross) -> ga,
// adj-select + lrelu -> homo alpha (ah), masked lrelu -> hete scores (sh).
// ---------------------------------------------------------------------------
__global__ __launch_bounds__(256) void k_scores(const float* __restrict__ q,
                                                const int* __restrict__ adj,
                                                const float* __restrict__ aParam,
                                                float* __restrict__ ga,
                                                float* __restrict__ ah,
                                                float* __restrict__ sh) {
    extern __shared__ float qs[];        // 128*128 floats (64 KB dynamic)
    __shared__ float aLds[4 * Dd];       // the 4 edge-attention vectors
    const int b = blockIdx.x;

    {
        const char* src = (const char*)(q + (size_t)b * Nn * Dd);
        for (int i = threadIdx.x; i < Nn * Dd * 4 / 16; i += 256)
            asyncCopy16((char*)qs + i * 16, src + (size_t)i * 16);
        for (int i = threadIdx.x; i < 4 * Dd * 4 / 16; i += 256)
            asyncCopy16((char*)aLds + i * 16, (const char*)aParam + (size_t)i * 16);
        asyncWait0();
    }
    __syncthreads();

    const int wave = threadIdx.x >> 5;
    const int lane = threadIdx.x & 31;
    const int nsel = lane & 15;
    const int hi   = lane >> 4;
    const int i0   = wave * 16;

    for (int s = 0; s < 8; ++s) {
        const int j0 = s * 16;
        v8f acc0 = {}, acc1 = {}, acc2 = {}, acc3 = {}, acc4 = {};
#pragma unroll 4
        for (int kk = 0; kk < 32; ++kk) {
            const int kb = kk * 4 + hi * 2;
            const float a0 = qs[(i0 + nsel) * Dd + kb];
            const float a1 = qs[(i0 + nsel) * Dd + kb + 1];
            v2f bf; bf.x = qs[(j0 + nsel) * Dd + kb];
                    bf.y = qs[(j0 + nsel) * Dd + kb + 1];
            v2f af; af.x = a0; af.y = a1;
            acc0 = wmma4(af, bf, acc0);
            v2f ar;
            ar.x = a0 * aLds[0 * Dd + kb]; ar.y = a1 * aLds[0 * Dd + kb + 1];
            acc1 = wmma4(ar, bf, acc1);
            ar.x = a0 * aLds[1 * Dd + kb]; ar.y = a1 * aLds[1 * Dd + kb + 1];
            acc2 = wmma4(ar, bf, acc2);
            ar.x = a0 * aLds[2 * Dd + kb]; ar.y = a1 * aLds[2 * Dd + kb + 1];
            acc3 = wmma4(ar, bf, acc3);
            ar.x = a0 * aLds[3 * Dd + kb]; ar.y = a1 * aLds[3 * Dd + kb + 1];
            acc4 = wmma4(ar, bf, acc4);
        }
#pragma unroll
        for (int v = 0; v < 8; ++v) {
            const int i = i0 + v + hi * 8;
            const int j = j0 + nsel;
            const size_t idx = ((size_t)b * Nn + i) * Nn + j;
            const int e = adj[idx];
            const float sc = acc0[v];
            ga[idx] = tanhf(sc);
            const float eh = (e == 1) ? acc1[v]
                           : (e == 2) ? acc2[v]
                           : (e == 3) ? acc3[v] : acc4[v];
            ah[idx] = (e >= 1 && e <= 4) ? lrelu(eh) : NEGV;
            const float hs = (e == 1 || (e >= 5 && e <= 10)) ? sc : NEGV;
            sh[idx] = lrelu(hs);
        }
    }
}

// ---------------------------------------------------------------------------
// Kernel 3: row softmax over ah and sh in place.  One wave32 per 128-elem
// row (4 floats/lane), shfl_xor reductions.
// ---------------------------------------------------------------------------
__global__ __launch_bounds__(256) void k_softmax(float* __restrict__ ah,
                                                 float* __restrict__ sh) {
    const int wave = threadIdx.x >> 5;
    const int lane = threadIdx.x & 31;
    const int rid = blockIdx.x * 8 + wave;          // 0 .. 2*B*N-1
    float* base = (rid < Bb * Nn) ? (ah + (size_t)rid * Nn)
                                  : (sh + (size_t)(rid - Bb * Nn) * Nn);
    float4 v = ((float4*)base)[lane];
    float mx = fmaxf(fmaxf(v.x, v.y), fmaxf(v.z, v.w));
#pragma unroll
    for (int off = 16; off > 0; off >>= 1) mx = fmaxf(mx, __shfl_xor(mx, off, 32));
    float e0 = __expf(v.x - mx), e1 = __expf(v.y - mx);
    float e2 = __expf(v.z - mx), e3 = __expf(v.w - mx);
    float sum = e0 + e1 + e2 + e3;
#pragma unroll
    for (int off = 16; off > 0; off >>= 1) sum += __shfl_xor(sum, off, 32);
    const float inv = 1.f / sum;
    float4 o; o.x = e0 * inv; o.y = e1 * inv; o.z = e2 * inv; o.w = e3 * inv;
    ((float4*)base)[lane] = o;
}

// ---------------------------------------------------------------------------
// Kernel 4: the three attention outputs, one block per batch.
// homo = ah @ q ; hete = sh @ q ; global = ga @ q  (shared B operand in LDS).
// ---------------------------------------------------------------------------
__global__ __launch_bounds__(256) void k_attnout(const float* __restrict__ q,
                                                 const float* __restrict__ ga,
                                                 const float* __restrict__ ah,
                                                 const float* __restrict__ sh,
                                                 float* __restrict__ ho,
                                                 float* __restrict__ he,
                                                 float* __restrict__ go) {
    extern __shared__ float qs[];   // 64 KB
    const int b = blockIdx.x;
    {
        const char* src = (const char*)(q + (size_t)b * Nn * Dd);
        for (int i = threadIdx.x; i < Nn * Dd * 4 / 16; i += 256)
            asyncCopy16((char*)qs + i * 16, src + (size_t)i * 16);
        asyncWait0();
    }
    __syncthreads();

    const int wave = threadIdx.x >> 5;
    const int lane = threadIdx.x & 31;
    const int nsel = lane & 15;
    const int hi   = lane >> 4;
    const int i0   = wave * 16;

    const float* A0 = ah + (size_t)b * Nn * Nn;
    const float* A1 = sh + (size_t)b * Nn * Nn;
    const float* A2 = ga + (size_t)b * Nn * Nn;

    for (int dt = 0; dt < 8; ++dt) {
        const int d0 = dt * 16;
        v8f acc0 = {}, acc1 = {}, acc2 = {};
#pragma unroll 4
        for (int kk = 0; kk < 32; ++kk) {
            const int kb = kk * 4 + hi * 2;
            v2f bf; bf.x = qs[kb * Dd + d0 + nsel];
                    bf.y = qs[(kb + 1) * Dd + d0 + nsel];
            v2f af;
            af.x = A0[(i0 + nsel) * Nn + kb]; af.y = A0[(i0 + nsel) * Nn + kb + 1];
            acc0 = wmma4(af, bf, acc0);
            af.x = A1[(i0 + nsel) * Nn + kb]; af.y = A1[(i0 + nsel) * Nn + kb + 1];
            acc1 = wmma4(af, bf, acc1);
            af.x = A2[(i0 + nsel) * Nn + kb]; af.y = A2[(i0 + nsel) * Nn + kb + 1];
            acc2 = wmma4(af, bf, acc2);
        }
#pragma unroll
        for (int v = 0; v < 8; ++v) {
            const int i = i0 + v + hi * 8;
            const size_t idx = ((size_t)b * Nn + i) * Dd + d0 + nsel;
            ho[idx] = acc0[v];
            he[idx] = acc1[v];
            go[idx] = acc2[v];
        }
    }
}

// ---------------------------------------------------------------------------
// Kernel 5: gate fusion, one block per batch.  Two MLP-energy passes done as
// WMMA GEMMs over LDS-resident matrices; per-tile relu+w2 dot reduced with
// shfl_xor and accumulated with LDS float atomics (ds_add_f32).
// ---------------------------------------------------------------------------
__global__ __launch_bounds__(256) void k_gate(const float* __restrict__ ho,
                                              const float* __restrict__ he,
                                              const float* __restrict__ go,
                                              const float* __restrict__ pw1,
                                              const float* __restrict__ pb1,
                                              const float* __restrict__ pw2,
                                              const float* __restrict__ pb2,
                                              const float* __restrict__ gw1,
                                              const float* __restrict__ gb1,
                                              const float* __restrict__ gw2,
                                              const float* __restrict__ gb2,
                                              float* __restrict__ out) {
    extern __shared__ float smem[];       // 2 * 64 KB dynamic
    float* M0 = smem;
    float* M1 = smem + Nn * Dd;
    __shared__ float eArr[2][Nn];

    const int b = blockIdx.x;
    const int tid = threadIdx.x;
    const int wave = tid >> 5;
    const int lane = tid & 31;
    const int nsel = lane & 15;
    const int hi   = lane >> 4;

    {
        const char* h4 = (const char*)(ho + (size_t)b * Nn * Dd);
        const char* t4 = (const char*)(he + (size_t)b * Nn * Dd);
        for (int i = tid; i < Nn * Dd * 4 / 16; i += 256) {
            asyncCopy16((char*)M0 + i * 16, h4 + (size_t)i * 16);
            asyncCopy16((char*)M1 + i * 16, t4 + (size_t)i * 16);
        }
        for (int i = tid; i < 2 * Nn; i += 256) ((float*)eArr)[i] = 0.f;
        asyncWait0();
    }
    __syncthreads();

    auto energyPass = [&](const float* w1, const float* b1, const float* w2) {
        for (int t = wave; t < 128; t += 8) {      // 2 matrices * 64 tiles
            const int m  = t >> 6;
            const int rm = t & 63;
            const int i0 = (rm >> 3) * 16;
            const int c0 = (rm & 7) * 16;
            const float* X = m ? M1 : M0;
            v8f acc = {};
#pragma unroll 8
            for (int kk = 0; kk < 32; ++kk) {
                const int kb = kk * 4 + hi * 2;
                v2f af; af.x = X[(i0 + nsel) * Dd + kb];
                        af.y = X[(i0 + nsel) * Dd + kb + 1];
                v2f bf; bf.x = w1[(size_t)kb * Dd + c0 + nsel];
                        bf.y = w1[(size_t)(kb + 1) * Dd + c0 + nsel];
                acc = wmma4(af, bf, acc);
            }
            const int col = c0 + nsel;
            const float bias = b1[col];
            const float wv = w2[col];
#pragma unroll
            for (int v = 0; v < 8; ++v) {
                float h = fmaxf(acc[v] + bias, 0.f);
                float c = h * wv;
                c += __shfl_xor(c, 1, 16);
                c += __shfl_xor(c, 2, 16);
                c += __shfl_xor(c, 4, 16);
                c += __shfl_xor(c, 8, 16);
                if (nsel == 0) atomicAdd(&eArr[m][i0 + v + hi * 8], c);
            }
        }
    };

    // ---- stage 1: homo vs hete ----
    energyPass(pw1, pb1, pw2);
    __syncthreads();
    {
        const float b2v = pb2[0];
        for (int idx = tid; idx < Nn * Dd; idx += 256) {
            const int n = idx >> 7;
            const float e0 = eArr[0][n] + b2v, e1 = eArr[1][n] + b2v;
            const float mx = fmaxf(e0, e1);
            const float x0 = __expf(e0 - mx), x1 = __expf(e1 - mx);
            const float inv = 1.f / (x0 + x1);
            M0[idx] = (x0 * M0[idx] + x1 * M1[idx]) * inv;   // local_output
        }
    }
    __syncthreads();
    {
        const char* g4 = (const char*)(go + (size_t)b * Nn * Dd);
        for (int i = tid; i < Nn * Dd * 4 / 16; i += 256)
            asyncCopy16((char*)M1 + i * 16, g4 + (size_t)i * 16);
        for (int i = tid; i < 2 * Nn; i += 256) ((float*)eArr)[i] = 0.f;
        asyncWait0();
    }
    __syncthreads();

    // ---- stage 2: local vs global ----
    energyPass(gw1, gb1, gw2);
    __syncthreads();
    {
        const float b2v = gb2[0];
        float* ob = out + (size_t)b * Nn * Dd;
        for (int idx = tid; idx < Nn * Dd; idx += 256) {
            const int n = idx >> 7;
            const float e0 = eArr[0][n] + b2v, e1 = eArr[1][n] + b2v;
            const float mx = fmaxf(e0, e1);
            const float x0 = __expf(e0 - mx), x1 = __expf(e1 - mx);
            const float inv = 1.f / (x0 + x1);
            ob[idx] = (x0 * M0[idx] + x1 * M1[idx]) * inv;
        }
    }
}

// ---------------------------------------------------------------------------
// ws layout (floats)
// ---------------------------------------------------------------------------
#define OFF_Q  ((size_t)0)
#define OFF_GA ((size_t)(Bb * Nn * Dd))
#define OFF_AH (OFF_GA + (size_t)Bb * Nn * Nn)
#define OFF_SH (OFF_AH + (size_t)Bb * Nn * Nn)
#define OFF_HO (OFF_SH + (size_t)Bb * Nn * Nn)
#define OFF_HE (OFF_HO + (size_t)Bb * Nn * Dd)
#define OFF_GO (OFF_HE + (size_t)Bb * Nn * Dd)

extern "C" void kernel_launch(void* const* d_in, const int* in_sizes, int n_in,
                              void* d_out, int out_size, void* d_ws, size_t ws_size,
                              hipStream_t stream) {
    const float* hidden = (const float*)d_in[0];
    const int*   adj    = (const int*)d_in[1];
    const int*   ntm    = (const int*)d_in[2];
    const float* WQ     = (const float*)d_in[3];
    // d_in[4] = W_K, d_in[5] = W_V : unused (reference bug preserved: k=v=q)
    const float* aParam = (const float*)d_in[6];
    const float* pw1 = (const float*)d_in[7];
    const float* pb1 = (const float*)d_in[8];
    const float* pw2 = (const float*)d_in[9];
    const float* pb2 = (const float*)d_in[10];
    const float* gw1 = (const float*)d_in[11];
    const float* gb1 = (const float*)d_in[12];
    const float* gw2 = (const float*)d_in[13];
    const float* gb2 = (const float*)d_in[14];

    float* ws  = (float*)d_ws;
    float* out = (float*)d_out;

    k_qproj<<<dim3(128), dim3(256), 0, stream>>>(hidden, ntm, WQ, ws + OFF_Q);

    k_scores<<<dim3(Bb), dim3(256), Nn * Dd * sizeof(float), stream>>>(
        ws + OFF_Q, adj, aParam, ws + OFF_GA, ws + OFF_AH, ws + OFF_SH);

    k_softmax<<<dim3(2 * Bb * Nn / 8), dim3(256), 0, stream>>>(ws + OFF_AH,
                                                               ws + OFF_SH);

    k_attnout<<<dim3(Bb), dim3(256), Nn * Dd * sizeof(float), stream>>>(
        ws + OFF_Q, ws + OFF_GA, ws + OFF_AH, ws + OFF_SH,
        ws + OFF_HO, ws + OFF_HE, ws + OFF_GO);

    k_gate<<<dim3(Bb), dim3(256), 2 * Nn * Dd * sizeof(float), stream>>>(
        ws + OFF_HO, ws + OFF_HE, ws + OFF_GO,
        pw1, pb1, pw2, pb2, gw1, gb1, gw2, gb2, out);
}